// LocalEncoder_71184787964088
// MI455X (gfx1250) — compile-verified
//
#include <hip/hip_runtime.h>
#include <math.h>

typedef __attribute__((ext_vector_type(2))) float v2f;
typedef __attribute__((ext_vector_type(8))) float v8f;

#define HID 64
#define NEG_SLOPE 0.2f

// ---------- float atomic max via integer atomics (always lowers to global_atomic_{max_i32,min_u32}) ----------
__device__ __forceinline__ void atomicMaxF(float* addr, float v) {
  if (v >= 0.0f)
    atomicMax((int*)addr, __float_as_int(v));
  else
    atomicMin((unsigned int*)addr, __float_as_uint(v));
}

// ---------- H = X * W  (X:[n,64] f32 row-major, W:[64,64] f32 row-major) via V_WMMA_F32_16X16X4_F32 ----------
// One wave computes a 16x64 row block. K swept in steps of 4. W staged in LDS (16 KB).
__global__ __launch_bounds__(128) void gemm64_wmma(const float* __restrict__ X,
                                                   const float* __restrict__ W,
                                                   float* __restrict__ H, int n) {
  __shared__ float sW[HID * HID];
  for (int i = threadIdx.x; i < HID * HID; i += 128) sW[i] = W[i];
  __syncthreads();

  const int wave = threadIdx.x >> 5;
  const int lane = threadIdx.x & 31;
  const int m0 = (blockIdx.x * 4 + wave) * 16;
  if (m0 >= n) return;                 // wave-uniform: EXEC stays all-ones for WMMA

  const int half = lane >> 4;          // 0: K pair {0,1}; 1: K pair {2,3} of each 16x4 A tile
  const int lr = lane & 15;            // M row within tile (A) / N column (B, C)

  v8f acc0 = {}, acc1 = {}, acc2 = {}, acc3 = {};
  const float* xrow = X + (size_t)(m0 + lr) * HID;

  for (int k = 0; k < HID; k += 4) {
    const int ka = k + 2 * half;
    v2f a;
    a.x = xrow[ka];
    a.y = xrow[ka + 1];
    const float* w0 = &sW[ka * HID + lr];
    const float* w1 = &sW[(ka + 1) * HID + lr];
    v2f b0, b1, b2, b3;
    b0.x = w0[0];  b0.y = w1[0];       // N tile 0  (cols  0..15)
    b1.x = w0[16]; b1.y = w1[16];      // N tile 1  (cols 16..31)
    b2.x = w0[32]; b2.y = w1[32];      // N tile 2  (cols 32..47)
    b3.x = w0[48]; b3.y = w1[48];      // N tile 3  (cols 48..63)
    acc0 = __builtin_amdgcn_wmma_f32_16x16x4_f32(false, a, false, b0, (short)0, acc0, false, false);
    acc1 = __builtin_amdgcn_wmma_f32_16x16x4_f32(false, a, false, b1, (short)0, acc1, false, false);
    acc2 = __builtin_amdgcn_wmma_f32_16x16x4_f32(false, a, false, b2, (short)0, acc2, false, false);
    acc3 = __builtin_amdgcn_wmma_f32_16x16x4_f32(false, a, false, b3, (short)0, acc3, false, false);
  }

  // C/D layout: VGPR v holds M=v (lanes 0-15) and M=v+8 (lanes 16-31), N = lane&15
#pragma unroll
  for (int v = 0; v < 8; ++v) {
    const int row = m0 + v + 8 * half;
    float* out = H + (size_t)row * HID + lr;
    out[0]  = acc0[v];
    out[16] = acc1[v];
    out[32] = acc2[v];
    out[48] = acc3[v];
  }
}

// ---------- per-node attention dots: as[i] = h[i].a_src ; ad[i] = h[i].a_dst (one wave per node) ----------
__global__ __launch_bounds__(256) void node_alpha(const float* __restrict__ H,
                                                  const float* __restrict__ a_s,
                                                  const float* __restrict__ a_d,
                                                  float* __restrict__ outs,
                                                  float* __restrict__ outd, int n) {
  const int node = (blockIdx.x * blockDim.x + threadIdx.x) >> 5;
  const int lane = threadIdx.x & 31;
  if (node >= n) return;
  const float* h = H + (size_t)node * HID;
  const float h0 = h[lane], h1 = h[lane + 32];
  float s = h0 * a_s[lane] + h1 * a_s[lane + 32];
  float d = h0 * a_d[lane] + h1 * a_d[lane + 32];
#pragma unroll
  for (int off = 16; off > 0; off >>= 1) {
    s += __shfl_xor(s, off, 32);
    d += __shfl_xor(d, off, 32);
  }
  if (lane == 0) { outs[node] = s; outd[node] = d; }
}

// ---------- fill ----------
__global__ void fill_kernel(float* __restrict__ p, float v, int n) {
  const int i = blockIdx.x * blockDim.x + threadIdx.x;
  if (i < n) p[i] = v;
}

// ---------- edge pass 1: logits + segment max over dst ----------
__global__ __launch_bounds__(256) void edge_logit_max(const int* __restrict__ src,
                                                      const int* __restrict__ dst,
                                                      const float* __restrict__ as_,
                                                      const float* __restrict__ ad_,
                                                      float* __restrict__ logits,
                                                      float* __restrict__ segmax,
                                                      int E, int ET) {
  const int e = blockIdx.x * blockDim.x + threadIdx.x;
  if (e >= ET) return;
  int s, d;
  if (e < E) { s = src[e]; d = dst[e]; } else { s = d = e - E; }  // implicit self-loops
  float l = as_[s] + ad_[d];
  l = (l >= 0.0f) ? l : NEG_SLOPE * l;
  logits[e] = l;
  atomicMaxF(&segmax[d], l);
}

// ---------- edge pass 2: ex = exp(l - max[dst]); denom[dst] += ex (ex stored in place of logits) ----------
__global__ __launch_bounds__(256) void edge_exp_sum(const int* __restrict__ dst,
                                                    float* __restrict__ logits,
                                                    const float* __restrict__ segmax,
                                                    float* __restrict__ denom,
                                                    int E, int ET) {
  const int e = blockIdx.x * blockDim.x + threadIdx.x;
  if (e >= ET) return;
  const int d = (e < E) ? dst[e] : (e - E);
  const float ex = __expf(logits[e] - segmax[d]);
  logits[e] = ex;
  atomicAdd(&denom[d], ex);
}

// ---------- edge pass 3: agg[dst] += (ex/denom[dst]) * h[src]  (one wave per edge, 2 channels/lane) ----------
__global__ __launch_bounds__(256) void edge_aggregate(const int* __restrict__ src,
                                                      const int* __restrict__ dst,
                                                      const float* __restrict__ H,
                                                      const float* __restrict__ ex,
                                                      const float* __restrict__ denom,
                                                      float* __restrict__ agg,
                                                      int E, int ET) {
  const int t = blockIdx.x * blockDim.x + threadIdx.x;
  const int e = t >> 5;
  const int lane = t & 31;
  if (e >= ET) return;
  int s, d;
  if (e < E) { s = src[e]; d = dst[e]; } else { s = d = e - E; }
  const float w = ex[e] / fmaxf(denom[d], 1e-16f);
  const float* h = H + (size_t)s * HID;
  atomicAdd(&agg[(size_t)d * HID + lane],      w * h[lane]);
  atomicAdd(&agg[(size_t)d * HID + lane + 32], w * h[lane + 32]);
}

// ---------- node finalize: feat = elu(agg + bias) ----------
__global__ __launch_bounds__(256) void node_finalize(const float* __restrict__ agg,
                                                     const float* __restrict__ bias,
                                                     float* __restrict__ feat, int total) {
  const int i = blockIdx.x * blockDim.x + threadIdx.x;
  if (i >= total) return;
  float v = agg[i] + bias[i & (HID - 1)];
  v = (v > 0.0f) ? v : (__expf(v) - 1.0f);
  feat[i] = v;
}

// ---------- pooling ----------
__global__ __launch_bounds__(256) void pool_accum(const float* __restrict__ feat,
                                                  const int* __restrict__ batch,
                                                  float* __restrict__ pool,
                                                  float* __restrict__ cnt, int total) {
  const int i = blockIdx.x * blockDim.x + threadIdx.x;
  if (i >= total) return;
  const int node = i >> 6;
  const int c = i & (HID - 1);
  const int g = batch[node];
  atomicAdd(&pool[(size_t)g * HID + c], feat[i]);
  if (c == 0) atomicAdd(&cnt[g], 1.0f);
}

__global__ __launch_bounds__(256) void pool_divide(const float* __restrict__ pool,
                                                   const float* __restrict__ cnt,
                                                   float* __restrict__ out, int total) {
  const int i = blockIdx.x * blockDim.x + threadIdx.x;
  if (i >= total) return;
  out[i] = pool[i] / fmaxf(cnt[i >> 6], 1.0f);
}

// =====================================================================================================

extern "C" void kernel_launch(void* const* d_in, const int* in_sizes, int n_in,
                              void* d_out, int out_size, void* d_ws, size_t ws_size,
                              hipStream_t stream) {
  const float* x      = (const float*)d_in[0];
  const int*   eidx   = (const int*)d_in[1];   // [2, E] flattened: src then dst
  const int*   batch  = (const int*)d_in[2];
  const float* W1     = (const float*)d_in[3];
  const float* a_src1 = (const float*)d_in[4];
  const float* a_dst1 = (const float*)d_in[5];
  const float* b1     = (const float*)d_in[6];
  const float* W2     = (const float*)d_in[7];
  const float* a_src2 = (const float*)d_in[8];
  const float* a_dst2 = (const float*)d_in[9];
  const float* b2     = (const float*)d_in[10];
  float* out = (float*)d_out;

  const int N  = in_sizes[0] / HID;       // 50000
  const int E  = in_sizes[1] / 2;         // 800000
  const int ET = E + N;                   // edges + self loops
  const int G  = out_size / HID;          // 512
  const int* src = eidx;
  const int* dst = eidx + E;

  // workspace layout (f32)
  float* ws     = (float*)d_ws;
  float* hbuf   = ws;                          // N*64  transformed features h
  float* feat   = hbuf + (size_t)N * HID;      // N*64  layer output (elu)
  float* agg    = feat + (size_t)N * HID;      // N*64  scatter accumulator
  float* as_    = agg  + (size_t)N * HID;      // N
  float* ad_    = as_ + N;                     // N
  float* segmax = ad_ + N;                     // N
  float* denom  = segmax + N;                  // N
  float* pool   = denom + N;                   // G*64
  float* cnt    = pool + (size_t)G * HID;      // G
  float* logits = cnt + G;                     // ET (reused as ex)

  const int gemm_blocks  = (N / 16 + 3) / 4;          // 16-row tiles, 4 waves/block
  const int alpha_blocks = (N * 32 + 255) / 256;
  const int edge_blocks  = (ET + 255) / 256;
  const int eagg_blocks  = (ET * 32 + 255) / 256;
  const int node_blocks  = (N * HID + 255) / 256;

  const float* in_feat = x;
  const float* Ws[2]     = {W1, W2};
  const float* As[2]     = {a_src1, a_src2};
  const float* Ad[2]     = {a_dst1, a_dst2};
  const float* Bs[2]     = {b1, b2};

  for (int layer = 0; layer < 2; ++layer) {
    gemm64_wmma<<<gemm_blocks, 128, 0, stream>>>(in_feat, Ws[layer], hbuf, N);
    node_alpha<<<alpha_blocks, 256, 0, stream>>>(hbuf, As[layer], Ad[layer], as_, ad_, N);

    fill_kernel<<<(N + 255) / 256, 256, 0, stream>>>(segmax, -3.0e38f, N);
    fill_kernel<<<(N + 255) / 256, 256, 0, stream>>>(denom, 0.0f, N);
    fill_kernel<<<node_blocks, 256, 0, stream>>>(agg, 0.0f, N * HID);

    edge_logit_max<<<edge_blocks, 256, 0, stream>>>(src, dst, as_, ad_, logits, segmax, E, ET);
    edge_exp_sum<<<edge_blocks, 256, 0, stream>>>(dst, logits, segmax, denom, E, ET);
    edge_aggregate<<<eagg_blocks, 256, 0, stream>>>(src, dst, hbuf, logits, denom, agg, E, ET);

    node_finalize<<<node_blocks, 256, 0, stream>>>(agg, Bs[layer], feat, N * HID);
    in_feat = feat;
  }

  fill_kernel<<<(G * HID + 255) / 256, 256, 0, stream>>>(pool, 0.0f, G * HID);
  fill_kernel<<<(G + 255) / 256, 256, 0, stream>>>(cnt, 0.0f, G);
  pool_accum<<<node_blocks, 256, 0, stream>>>(feat, batch, pool, cnt, N * HID);
  pool_divide<<<(G * HID + 255) / 256, 256, 0, stream>>>(pool, cnt, out, G * HID);
}